// LocalAttentionCache_65506841198820
// MI455X (gfx1250) — compile-verified
//
#include <hip/hip_runtime.h>
#include <hip/hip_bf16.h>
#include <math.h>

// ---------------------------------------------------------------------------
// kNN (k=32) + polar relative positional encoding for B=4, L=4096 2-D points.
// dist^2 computed via V_WMMA_F32_16X16X4_F32 Gram tiles:
//   dist2[m][n] = |q_m|^2 + |c_n|^2 - 2 * (q_m . c_n)
// WMMA result used as a filtered pre-screen; exact recompute near threshold.
// ---------------------------------------------------------------------------

typedef float v2f __attribute__((ext_vector_type(2)));
typedef float v8f __attribute__((ext_vector_type(8)));

#define L_FIXED 4096
#define KNN     32
#define NFREQ   16
#define CHUNK   512
#define TILES_PER_CHUNK (CHUNK / 16)
#define PI_F    3.14159265358979323846f

__global__ __launch_bounds__(32) void knn_polar_rpe_kernel(
    const float* __restrict__ pos,   // [B, L, 2]
    float* __restrict__ out,         // idx | rpe | dist, concatenated flat
    int B, int L)
{
    __shared__ float chunkbuf[16 * CHUNK];   // dist^2 tile buffer (32 KB)
    __shared__ float cpos[CHUNK * 2];        // chunk candidate coords (4 KB)
    __shared__ float topd[16][KNN];          // per-query sorted best dist^2
    __shared__ int   topi[16][KNN];          // per-query best indices
    __shared__ float qpx[16], qpy[16], qns[16];

    const int lane  = threadIdx.x;           // 0..31 (wave32)
    const int b     = blockIdx.y;
    const int qbase = blockIdx.x * 16;       // query tile start
    const int hi    = lane >> 4;             // half-wave
    const int nl    = lane & 15;

    const float* bpos = pos + (size_t)b * L * 2;

    // ---- load 16 query points, squared norms ----
    if (lane < 16) {
        float qx = bpos[(qbase + lane) * 2 + 0];
        float qy = bpos[(qbase + lane) * 2 + 1];
        qpx[lane] = qx;
        qpy[lane] = qy;
        qns[lane] = qx * qx + qy * qy;
    }
    // init top-k lists
    for (int j = lane; j < 16 * KNN; j += 32) {
        (&topd[0][0])[j] = 3.0e38f;
        (&topi[0][0])[j] = 0;
    }
    __syncthreads();

    // ---- A matrix (16x4 f32): lanes 0-15 carry K0=x, K1=y; lanes 16-31 pad 0
    v2f amat;
    amat.x = (hi == 0) ? qpx[nl] : 0.0f;
    amat.y = (hi == 0) ? qpy[nl] : 0.0f;

    // per-lane query norms for this lane's 8 C rows (M = v + 8*hi)
    float qn8[8];
#pragma unroll
    for (int v = 0; v < 8; ++v) qn8[v] = qns[v + 8 * hi];

    // ---- candidate chunks ----
    for (int ch = 0; ch < L; ch += CHUNK) {
        // build + WMMA: 32 tiles of 16 candidates (full wave, EXEC all-ones)
        for (int t = 0; t < TILES_PER_CHUNK; ++t) {
            const int nn   = t * 16 + nl;        // candidate within chunk
            const int cidx = ch + nn;
            const float cx = bpos[cidx * 2 + 0]; // both halves load same cand
            const float cy = bpos[cidx * 2 + 1];
            cpos[nn * 2 + 0] = cx;               // duplicate writes, same value
            cpos[nn * 2 + 1] = cy;
            const float cn = cx * cx + cy * cy;

            // B matrix (4x16 f32): lanes 0-15 carry K0=x, K1=y; pad rows zero
            v2f bmat;
            bmat.x = (hi == 0) ? cx : 0.0f;
            bmat.y = (hi == 0) ? cy : 0.0f;

            v8f acc = {0.f, 0.f, 0.f, 0.f, 0.f, 0.f, 0.f, 0.f};
            acc = __builtin_amdgcn_wmma_f32_16x16x4_f32(
                /*neg_a=*/false, amat, /*neg_b=*/false, bmat,
                /*c_mod=*/(short)0, acc, /*reuse_a=*/false, /*reuse_b=*/false);

            // C layout: lane l, vgpr v -> M = v + 8*(l>=16), N = l & 15
#pragma unroll
            for (int v = 0; v < 8; ++v) {
                float d2 = qn8[v] + cn - 2.0f * acc[v];
                d2 = fmaxf(d2, 0.0f);
                chunkbuf[(v + 8 * hi) * CHUNK + nn] = d2;
            }
        }
        __syncthreads();

        // selection: lane q (q<16) owns query q; sorted insertion into top-32.
        // WMMA dist^2 has ~O(1) abs error -> filter with margin, verify exact.
        if (lane < 16) {
            const int q = lane;
            const float qx = qpx[q], qy = qpy[q];
            float worst = topd[q][KNN - 1];
            for (int nn = 0; nn < CHUNK; ++nn) {
                const float ad = chunkbuf[q * CHUNK + nn];
                if (ad < worst + 16.0f) {
                    const float dx = cpos[nn * 2 + 0] - qx;
                    const float dy = cpos[nn * 2 + 1] - qy;
                    const float d2 = dx * dx + dy * dy;   // exact
                    if (d2 < worst) {
                        int j = KNN - 1;
                        while (j > 0 && topd[q][j - 1] > d2) {
                            topd[q][j] = topd[q][j - 1];
                            topi[q][j] = topi[q][j - 1];
                            --j;
                        }
                        topd[q][j] = d2;
                        topi[q][j] = ch + nn;
                        worst = topd[q][KNN - 1];
                    }
                }
            }
        }
        __syncthreads();
    }

    // ---- outputs: 16 queries x 32 neighbors = 512 pairs over 32 lanes ----
    const float physical_scale = 3.0f * sqrtf((float)KNN / PI_F);
    const float inv_scale = 1.0f / physical_scale;
    const size_t BLK = (size_t)B * L * KNN;
    int*   idx_out  = (int*)out;               // region 0: indices (int32 bits)
    float* rpe_out  = out + BLK;               // region 1: [B,L,k,64]
    float* dist_out = out + BLK + BLK * 64;    // region 2: distances

    for (int p = lane; p < 16 * KNN; p += 32) {
        const int q  = p / KNN;
        const int j  = p % KNN;
        const int qg = qbase + q;
        const int idx = topi[q][j];
        const float qx = qpx[q], qy = qpy[q];
        const float dx = bpos[idx * 2 + 0] - qx;
        const float dy = bpos[idx * 2 + 1] - qy;
        const float dist = sqrtf(dx * dx + dy * dy + 1e-8f);
        const float r  = dist * inv_scale;
        const float th = atan2f(dy, dx);

        const size_t o = ((size_t)b * L + qg) * KNN + j;
        idx_out[o]  = idx;
        dist_out[o] = dist;

        float* rp = rpe_out + o * 64;
        float freq = PI_F;                     // 2^f * pi
#pragma unroll
        for (int f = 0; f < NFREQ; ++f) {
            const float rf = r * freq;
            const float tf = th * freq;
            rp[f +  0] = __sinf(rf);
            rp[f + 16] = __cosf(rf);
            rp[f + 32] = __sinf(tf);
            rp[f + 48] = __cosf(tf);
            freq *= 2.0f;
        }
    }
}

extern "C" void kernel_launch(void* const* d_in, const int* in_sizes, int n_in,
                              void* d_out, int out_size, void* d_ws, size_t ws_size,
                              hipStream_t stream) {
    const float* pos = (const float*)d_in[0];
    (void)d_ws; (void)ws_size; (void)n_in; (void)out_size;

    int L = L_FIXED;
    int B = in_sizes[0] / (2 * L);
    if (B < 1) { B = 1; L = in_sizes[0] / 2; }   // defensive fallback

    dim3 grid(L / 16, B);
    knn_polar_rpe_kernel<<<grid, 32, 0, stream>>>(pos, (float*)d_out, B, L);
}